// TemporalGATLayer_68607807586901
// MI455X (gfx1250) — compile-verified
//
#include <hip/hip_runtime.h>
#include <math.h>

// Temporal GAT layer for MI455X (gfx1250, wave32).
// Memory-bound scatter/gather problem (~2 GB traffic/call @ 23.3 TB/s);
// the lone GEMM (20000x256 @ 256x256) runs on V_WMMA_F32_16X16X4_F32.

#define IN_DIM   128
#define TIME_DIM 128
#define FEAT     256   // IN_DIM + TIME_DIM
#define HEADS    4
#define DHEAD    64
#define OUTF     256   // HEADS*DHEAD

typedef float v2f __attribute__((ext_vector_type(2)));
typedef float v8f __attribute__((ext_vector_type(8)));

// ---------------------------------------------------------------- utilities
__device__ __forceinline__ void atomicMaxF32(float* addr, float val) {
    // classic signed/unsigned bit trick; valid with -inf initialization
    if (val >= 0.0f) atomicMax((int*)addr, __float_as_int(val));
    else             atomicMin((unsigned int*)addr, __float_as_uint(val));
}

// ---------------------------------------------------------------- K0: init
// zero msum/deg/denom/out, set mbuf to -inf.  layout: [N*128 | N | 4N | 4N | 256N]
__global__ void k_init(float* __restrict__ msum, float* __restrict__ deg,
                       float* __restrict__ mbuf, float* __restrict__ denom,
                       float* __restrict__ out, int N) {
    const long a = (long)N * TIME_DIM;
    const long b = a + N;
    const long c = b + 4L * N;
    const long d = c + 4L * N;
    const long total = d + (long)N * OUTF;
    for (long i = (long)blockIdx.x * blockDim.x + threadIdx.x; i < total;
         i += (long)gridDim.x * blockDim.x) {
        if      (i < a) msum[i]      = 0.0f;
        else if (i < b) deg[i - a]   = 0.0f;
        else if (i < c) mbuf[i - b]  = -INFINITY;
        else if (i < d) denom[i - c] = 0.0f;
        else            out[i - d]   = 0.0f;
    }
}

// ---------------------------------------------------------- K1: time scatter
// one wave per edge; 32 lanes x 4 features.  msg = h[src] + cos(t*w+b),
// atomically accumulated into msum[dst]; deg[dst] += 1.
__global__ void k_time_scatter(const float* __restrict__ h, const float* __restrict__ t,
                               const float* __restrict__ tw, const float* __restrict__ tb,
                               const int* __restrict__ src, const int* __restrict__ dst,
                               float* __restrict__ msum, float* __restrict__ deg, int E) {
    int e    = blockIdx.x * 4 + (threadIdx.x >> 5);
    int lane = threadIdx.x & 31;
    if (e >= E) return;
    int s = src[e], d = dst[e];
    float tv = t[e];
    const float* hs = h + (size_t)s * IN_DIM;
    float*       ms = msum + (size_t)d * TIME_DIM;
#pragma unroll
    for (int i = 0; i < 4; ++i) {
        int f = lane + 32 * i;
        float te = cosf(tv * tw[f] + tb[f]);
        atomicAdd(&ms[f], hs[f] + te);
    }
    if (lane == 0) atomicAdd(&deg[d], 1.0f);
}

// ---------------------------------------------------------- K2: build h_wt
__global__ void k_build_hwt(const float* __restrict__ h, const float* __restrict__ msum,
                            const float* __restrict__ deg, float* __restrict__ hwt) {
    int n = blockIdx.x;
    int f = threadIdx.x;                       // 0..255
    float v;
    if (f < IN_DIM) {
        v = h[(size_t)n * IN_DIM + f];
    } else {
        float dg = deg[n];
        v = (dg > 0.0f) ? msum[(size_t)n * TIME_DIM + (f - IN_DIM)] / dg : 0.0f;
    }
    hwt[(size_t)n * FEAT + f] = v;
}

// ------------------------------------------------- K3: z = h_wt @ fc_w^T (WMMA)
// One wave computes a 16x16 tile of z.  V_WMMA_F32_16X16X4_F32, K stepped by 4.
// Lane layout (ISA 7.12.2):
//   A 16x4 : lane<16 holds K={kb,kb+1}, lane>=16 holds K={kb+2,kb+3}, row = lane%16
//   B 4x16 : symmetric, col = lane%16
//   C/D    : VGPR v -> row m0 + v + 8*(lane>=16), col = n0 + lane%16
__global__ void k_gemm_wmma(const float* __restrict__ hwt, const float* __restrict__ fcw,
                            float* __restrict__ z, int N) {
    int lane = threadIdx.x & 31;
    int half = lane >> 4;            // 0 | 1
    int idx  = lane & 15;
    int m0   = blockIdx.x * 16;
    int n0   = (blockIdx.y * blockDim.y + threadIdx.y) * 16;
    if (m0 >= N) return;             // uniform per block: EXEC stays all-ones

    const float* Arow = hwt + (size_t)(m0 + idx) * FEAT;       // A row-major
    const float* Brow = fcw + (size_t)(n0 + idx) * FEAT;       // B[k][n] = fc_w[n][k]
    const int ko = 2 * half;

    v8f acc = {};
#pragma unroll 4
    for (int kb = 0; kb < FEAT; kb += 4) {
        v2f a, b;
        a.x = Arow[kb + ko];  a.y = Arow[kb + ko + 1];
        b.x = Brow[kb + ko];  b.y = Brow[kb + ko + 1];
        acc = __builtin_amdgcn_wmma_f32_16x16x4_f32(
            /*neg_a=*/false, a, /*neg_b=*/false, b,
            /*c_mod=*/(short)0, acc, /*reuse_a=*/false, /*reuse_b=*/false);
    }

    int col = n0 + idx;
#pragma unroll
    for (int v = 0; v < 8; ++v) {
        int row = m0 + v + half * 8;
        z[(size_t)row * OUTF + col] = acc[v];
    }
}

// ------------------------------------------------- K4: el/er per (node, head)
__global__ void k_attn_dots(const float* __restrict__ z, const float* __restrict__ al,
                            const float* __restrict__ ar, float* __restrict__ el,
                            float* __restrict__ er, int N) {
    int t = blockIdx.x * blockDim.x + threadIdx.x;
    if (t >= N * HEADS) return;
    int n = t >> 2, hh = t & 3;
    const float* zp  = z  + (size_t)n * OUTF + hh * DHEAD;
    const float* alp = al + hh * DHEAD;
    const float* arp = ar + hh * DHEAD;
    float sl = 0.0f, sr = 0.0f;
#pragma unroll 8
    for (int d = 0; d < DHEAD; ++d) {
        float zv = zp[d];
        sl = fmaf(zv, alp[d], sl);
        sr = fmaf(zv, arp[d], sr);
    }
    el[t] = sl;
    er[t] = sr;
}

// ------------------------------------------------- K5: edge logits + segment max
__global__ void k_edge_logits(const float* __restrict__ el, const float* __restrict__ er,
                              const int* __restrict__ src, const int* __restrict__ dst,
                              float* __restrict__ ebuf, float* __restrict__ mbuf, int E) {
    int t = blockIdx.x * blockDim.x + threadIdx.x;
    if (t >= E * HEADS) return;
    int e = t >> 2, hh = t & 3;
    float v = el[src[e] * HEADS + hh] + er[dst[e] * HEADS + hh];
    v = (v > 0.0f) ? v : 0.2f * v;           // leaky_relu(0.2)
    ebuf[t] = v;
    atomicMaxF32(&mbuf[dst[e] * HEADS + hh], v);
}

// ------------------------------------------------- K6: exp + segment denom
__global__ void k_edge_exp(const float* __restrict__ mbuf, const int* __restrict__ dst,
                           float* __restrict__ ebuf, float* __restrict__ denom, int E) {
    int t = blockIdx.x * blockDim.x + threadIdx.x;
    if (t >= E * HEADS) return;
    int e = t >> 2, hh = t & 3;
    float ex = expf(ebuf[t] - mbuf[dst[e] * HEADS + hh]);
    ebuf[t] = ex;
    atomicAdd(&denom[dst[e] * HEADS + hh], ex);
}

// ------------------------------------------------- K7: rst += alpha * z[src]
// one wave per edge; 32 lanes x 8 features; accumulates directly into d_out
__global__ void k_aggregate(const float* __restrict__ z, const float* __restrict__ ebuf,
                            const float* __restrict__ denom, const int* __restrict__ src,
                            const int* __restrict__ dst, float* __restrict__ out, int E) {
    int e    = blockIdx.x * 4 + (threadIdx.x >> 5);
    int lane = threadIdx.x & 31;
    if (e >= E) return;
    int s = src[e], d = dst[e];
    const float* zs = z + (size_t)s * OUTF;
    float*       od = out + (size_t)d * OUTF;
    float alpha[HEADS];
#pragma unroll
    for (int hh = 0; hh < HEADS; ++hh)
        alpha[hh] = ebuf[e * HEADS + hh] / denom[d * HEADS + hh];
#pragma unroll
    for (int i = 0; i < 8; ++i) {
        int f = lane + 32 * i;
        atomicAdd(&od[f], alpha[f >> 6] * zs[f]);
    }
}

// ------------------------------------------------- K8: residual + bias + ELU
__global__ void k_finalize(const float* __restrict__ hwt, const float* __restrict__ bias,
                           float* __restrict__ out) {
    size_t t = (size_t)blockIdx.x * OUTF + threadIdx.x;
    float v = out[t] + hwt[t] + bias[threadIdx.x];
    out[t] = (v > 0.0f) ? v : expm1f(v);     // elu(alpha=1)
}

// ==================================================================== launch
extern "C" void kernel_launch(void* const* d_in, const int* in_sizes, int n_in,
                              void* d_out, int out_size, void* d_ws, size_t ws_size,
                              hipStream_t stream) {
    const float* h    = (const float*)d_in[0];
    const float* tt   = (const float*)d_in[1];
    const float* tw   = (const float*)d_in[2];
    const float* tb   = (const float*)d_in[3];
    const float* fcw  = (const float*)d_in[4];
    const float* al   = (const float*)d_in[5];
    const float* ar   = (const float*)d_in[6];
    const float* bias = (const float*)d_in[7];
    const int*   src  = (const int*)d_in[8];
    const int*   dst  = (const int*)d_in[9];
    float* out = (float*)d_out;

    const int N = in_sizes[0] / IN_DIM;
    const int E = in_sizes[1];

    // workspace carve-out (256B aligned)
    char* wsp = (char*)d_ws;
    auto carve = [&](size_t bytes) -> float* {
        float* p = (float*)wsp;
        wsp += (bytes + 255) & ~(size_t)255;
        return p;
    };
    float* msum  = carve((size_t)N * TIME_DIM * 4);   // segment sum of messages
    float* deg   = carve((size_t)N * 4);              // in-degree
    float* hwt   = carve((size_t)N * FEAT * 4);       // concat(h, h_time)
    float* zbuf  = carve((size_t)N * OUTF * 4);       // GEMM output
    float* el    = carve((size_t)N * HEADS * 4);
    float* er    = carve((size_t)N * HEADS * 4);
    float* mbuf  = carve((size_t)N * HEADS * 4);      // segment max
    float* denom = carve((size_t)N * HEADS * 4);      // segment sum of exp
    float* ebuf  = carve((size_t)E * HEADS * 4);      // logits -> exp(logits - max)

    // K0: init accumulators (also zeroes d_out, which doubles as rst accumulator)
    {
        long total = (long)N * (TIME_DIM + 1 + 4 + 4 + OUTF);
        int grid = (int)((total + 255) / 256);
        k_init<<<grid, 256, 0, stream>>>(msum, deg, mbuf, denom, out, N);
    }
    // K1: time encoding + scatter-mean numerator / denominator
    k_time_scatter<<<(E + 3) / 4, 128, 0, stream>>>(h, tt, tw, tb, src, dst, msum, deg, E);
    // K2: h_wt = concat(h, msum/deg)
    k_build_hwt<<<N, FEAT, 0, stream>>>(h, msum, deg, hwt);
    // K3: z = h_wt @ fc_w^T  (WMMA f32 16x16x4)
    {
        dim3 grid((N + 15) / 16, OUTF / 16 / 4);   // y: 4 tiles x blockDim.y(4) = 16 col-tiles
        dim3 block(32, 4);
        k_gemm_wmma<<<grid, block, 0, stream>>>(hwt, fcw, zbuf, N);
    }
    // K4: per-(node,head) attention dots
    k_attn_dots<<<(N * HEADS + 255) / 256, 256, 0, stream>>>(zbuf, al, ar, el, er, N);
    // K5: edge logits + segment max
    k_edge_logits<<<(E * HEADS + 255) / 256, 256, 0, stream>>>(el, er, src, dst, ebuf, mbuf, E);
    // K6: exp + segment denom
    k_edge_exp<<<(E * HEADS + 255) / 256, 256, 0, stream>>>(mbuf, dst, ebuf, denom, E);
    // K7: weighted aggregation into d_out
    k_aggregate<<<(E + 3) / 4, 128, 0, stream>>>(zbuf, ebuf, denom, src, dst, out, E);
    // K8: residual + bias + ELU (in place on d_out)
    k_finalize<<<N, OUTF, 0, stream>>>(hwt, bias, out);
}